// SimpleCFTLayer2d_44358422233766
// MI455X (gfx1250) — compile-verified
//
#include <hip/hip_runtime.h>

// Problem constants (fixed by the reference)
#define SEGN 4
#define BN   8
#define CN   64
#define ON   64
#define HN   512
#define WN   512
#define HS   (HN / SEGN)   // 128
#define WSV  (WN / SEGN)   // 128

typedef __attribute__((ext_vector_type(2))) float v2f;
typedef __attribute__((ext_vector_type(4))) float v4f;
typedef __attribute__((ext_vector_type(8))) float v8f;

// ---------------------------------------------------------------------------
// Kernel 1: segment mean-pool.  x[B,C,H,W] -> pooled[B,C,4,4]
// One workgroup per (b*C+c, i) band: 128 rows x 512 cols = 256 KiB streamed.
// Thread t always lands on within-row column (t&127)*4 (stride 1024 floats
// == 2 rows per iteration), so its segment column j = (t&127)/32 is FIXED.
// ---------------------------------------------------------------------------
__global__ void __launch_bounds__(256)
pool_kernel(const float* __restrict__ x, float* __restrict__ pooled) {
    const int blk = blockIdx.x;            // 0 .. B*C*SEG-1
    const int i   = blk & (SEGN - 1);      // segment row
    const int bc  = blk >> 2;              // b*C + c
    const int t   = threadIdx.x;           // 0..255

    const int col    = (t & 127) * 4;      // fixed within-row column
    const int rowpar = t >> 7;             // 0 or 1 (row parity)

    const float* src = x + ((size_t)bc * HN + (size_t)i * HS + rowpar) * WN + col;
    const v4f* p     = (const v4f*)src;

    float sum = 0.f;
    // 128 rows * 512 cols = 65536 floats; 1024 floats/iter -> 64 iters.
    #pragma unroll 4
    for (int it = 0; it < 64; ++it) {
        v4f v = __builtin_nontemporal_load(p + (size_t)it * 256); // +2 rows
        sum += (v.x + v.y) + (v.z + v.w);
    }

    __shared__ float red[256];
    red[t] = sum;
    __syncthreads();

    // Threads belonging to segment column j: t%128 in [j*32, j*32+32), both
    // row parities -> 64 partials per j.  Threads 0..3 finalize j=0..3.
    if (t < SEGN) {
        float s = 0.f;
        const int base = t * 32;
        #pragma unroll
        for (int k = 0; k < 32; ++k)
            s += red[base + k] + red[128 + base + k];
        // pooled flat index: (bc*4 + i)*4 + j  ==  [B,C,S,S] row-major
        pooled[(size_t)blk * SEGN + t] = s * (1.0f / (float)(HS * WSV));
    }
}

// ---------------------------------------------------------------------------
// Kernel 2: weighted[B,O,4,4] = (P[B*16,64] x W^T + bias) * seg_w
// via V_WMMA_F32_16X16X4_F32.  One wave per 16x16 output tile:
//   rowTile rt = batch b (16 rows = the 16 (i,j) cells), colTile ct = o/16.
// 16 chained WMMAs accumulate over C=64 in K=4 steps.
// ISA layouts (CDNA5 ISA):
//   A 16x4 f32 : lane L -> M = L&15 ; v0 = K=2*(L>>4), v1 = K=2*(L>>4)+1
//   B 4x16 f32 : lane L -> N = L&15 ; v0 = K=2*(L>>4), v1 = K=2*(L>>4)+1
//   C/D 16x16  : lane L -> N = L&15 ; vgpr v -> M = v + 8*(L>>4)
// ---------------------------------------------------------------------------
__global__ void __launch_bounds__(32)
gemm_kernel(const float* __restrict__ pooled,   // [B,C,4,4]
            const float* __restrict__ Wm,       // [O,C]
            const float* __restrict__ bias,     // [O]
            const float* __restrict__ segw,     // [4,4] flat 16
            float* __restrict__ weighted) {     // [B,O,4,4]
    const int tile = blockIdx.x;     // 0..31
    const int rt   = tile >> 2;      // batch b: 0..7
    const int ct   = tile & 3;       // o tile: 0..3

    const int lane  = threadIdx.x;   // 0..31
    const int mloc  = lane & 15;     // M for A, N for B/C/D
    const int khalf = lane >> 4;     // upper lanes carry K+2, K+3

    const int b  = rt;
    const int ij = mloc;                          // A-row -> (i,j) cell
    const int n  = ct * 16 + mloc;                // global o for B/C/D lanes

    v8f acc = {};
    #pragma unroll
    for (int k0 = 0; k0 < CN; k0 += 4) {
        const int ka = k0 + 2 * khalf;            // this lane's K, K+1
        v2f a, bm;
        // A: P[(b,ij)][c] = pooled[((b*C + c)*16) + ij]
        a.x = pooled[((size_t)(b * CN + ka    ) * 16) + ij];
        a.y = pooled[((size_t)(b * CN + ka + 1) * 16) + ij];
        // B: B[k][n] = W[n][k]  (W is [O,C] row-major)
        bm.x = Wm[(size_t)n * CN + ka];
        bm.y = Wm[(size_t)n * CN + ka + 1];
        acc = __builtin_amdgcn_wmma_f32_16x16x4_f32(
            /*neg_a=*/false, a, /*neg_b=*/false, bm,
            /*c_mod=*/(short)0, acc, /*reuse_a=*/false, /*reuse_b=*/false);
    }

    const float bn = bias[n];
    #pragma unroll
    for (int v = 0; v < 8; ++v) {
        const int mrow = v + 8 * khalf;           // (i,j) flat = i*4+j
        const float out = (acc[v] + bn) * segw[mrow];
        // weighted flat: ((b*O + o)*16) + (i*4+j)
        weighted[((size_t)(b * ON + n) * 16) + mrow] = out;
    }
}

// ---------------------------------------------------------------------------
// Kernel 3: broadcast each tile value over its 128x128 block.
// One workgroup per (b*O+o, i) band; same fixed-column pattern as kernel 1.
// Pure streaming NT stores: 512 MiB total.
// ---------------------------------------------------------------------------
__global__ void __launch_bounds__(256)
bcast_kernel(const float* __restrict__ weighted, float* __restrict__ out) {
    const int blk = blockIdx.x;            // 0 .. B*O*SEG-1
    const int i   = blk & (SEGN - 1);
    const int bo  = blk >> 2;              // b*O + o
    const int t   = threadIdx.x;

    const int col    = (t & 127) * 4;
    const int rowpar = t >> 7;
    const int j      = col >> 7;           // fixed segment column

    const float v = weighted[(size_t)bo * 16 + i * SEGN + j];
    v4f val;
    val.x = v; val.y = v; val.z = v; val.w = v;

    v4f* dst = (v4f*)(out + ((size_t)bo * HN + (size_t)i * HS + rowpar) * WN + col);
    #pragma unroll 4
    for (int it = 0; it < 64; ++it) {
        __builtin_nontemporal_store(val, dst + (size_t)it * 256);
    }
}

// ---------------------------------------------------------------------------
extern "C" void kernel_launch(void* const* d_in, const int* in_sizes, int n_in,
                              void* d_out, int out_size, void* d_ws, size_t ws_size,
                              hipStream_t stream) {
    (void)in_sizes; (void)n_in; (void)out_size; (void)ws_size;
    const float* x    = (const float*)d_in[0];   // [8,64,512,512]
    const float* Wm   = (const float*)d_in[1];   // [64,64]
    const float* bias = (const float*)d_in[2];   // [64]
    const float* segw = (const float*)d_in[3];   // [4,4]
    float* out = (float*)d_out;                  // [8,64,512,512]

    float* pooled   = (float*)d_ws;              // 8192 floats (32 KiB)
    float* weighted = (float*)d_ws + 8192;       // 8192 floats (32 KiB)

    pool_kernel <<<BN * CN * SEGN, 256, 0, stream>>>(x, pooled);
    gemm_kernel <<<32, 32, 0, stream>>>(pooled, Wm, bias, segw, weighted);
    bcast_kernel<<<BN * ON * SEGN, 256, 0, stream>>>(weighted, out);
}